// RenderLayer_29137058136659
// MI455X (gfx1250) — compile-verified
//
#include <hip/hip_runtime.h>

typedef float v2f __attribute__((ext_vector_type(2)));
typedef float v8f __attribute__((ext_vector_type(8)));

#define HH 512
#define WW 512
#define NIMG 2
#define NV 4096
#define NF 2048
#define TEXW 1024
#define NCHUNK 128          // NF / 16
#define EPSF 1e-8f
#define BIGF 1e30f

// ---------------- K1: project vertices -> v_pix (x,y,z,pad) ----------------
__global__ void k_project(const float* __restrict__ v, const float* __restrict__ campos,
                          const float* __restrict__ camrot, const float* __restrict__ focal,
                          const float* __restrict__ princpt, float4* __restrict__ vpix) {
    int t = blockIdx.x * blockDim.x + threadIdx.x;
    if (t >= NIMG * NV) return;
    int n = t / NV, vid = t % NV;
    const float* vv = v + (size_t)(n * NV + vid) * 3;
    float dx = vv[0] - campos[n * 3 + 0];
    float dy = vv[1] - campos[n * 3 + 1];
    float dz = vv[2] - campos[n * 3 + 2];
    const float* R = camrot + n * 9;
    float cx = R[0] * dx + R[1] * dy + R[2] * dz;
    float cy = R[3] * dx + R[4] * dy + R[5] * dz;
    float cz = R[6] * dx + R[7] * dy + R[8] * dz;
    float z = fmaxf(cz, EPSF);
    const float* Fm = focal + n * 4;
    float sx = cx / z, sy = cy / z;
    float xp = Fm[0] * sx + Fm[1] * sy + princpt[n * 2 + 0];
    float yp = Fm[2] * sx + Fm[3] * sy + princpt[n * 2 + 1];
    float4 o; o.x = xp; o.y = yp; o.z = cz; o.w = 0.f;
    vpix[n * NV + vid] = o;
}

// ---------------- K2: per-triangle setup -> WMMA B-matrix columns ----------------
// Layout: tri[n][chunk][func(4)][col(16)][4 floats]; func = e0,e1,e2,zi
__global__ void k_trisetup(const float4* __restrict__ vpix, const int* __restrict__ vi,
                           float* __restrict__ tri) {
    int t = blockIdx.x * blockDim.x + threadIdx.x;
    if (t >= NIMG * NF) return;
    int n = t / NF, f = t % NF;
    int i0 = vi[f * 3 + 0], i1 = vi[f * 3 + 1], i2 = vi[f * 3 + 2];
    float4 p0 = vpix[n * NV + i0];
    float4 p1 = vpix[n * NV + i1];
    float4 p2 = vpix[n * NV + i2];
    float x0 = p0.x, y0 = p0.y, z0 = p0.z;
    float x1 = p1.x, y1 = p1.y, z1 = p1.z;
    float x2 = p2.x, y2 = p2.y, z2 = p2.z;
    // w_k = A*px + B*py + C
    float A0 = y2 - y1, B0 = x1 - x2, C0 = x2 * y1 - x1 * y2;
    float A1 = y0 - y2, B1 = x2 - x0, C1 = x0 * y2 - x2 * y0;
    float A2 = y1 - y0, B2 = x0 - x1, C2 = x1 * y0 - x0 * y1;
    float area = (x1 - x0) * (y2 - y0) - (y1 - y0) * (x2 - x0);
    bool areaok = fabsf(area) > EPSF;
    bool zok = areaok && (z0 > EPSF) && (z1 > EPSF) && (z2 > EPSF);
    float sa = areaok ? area : 1.0f;
    float s = (area >= 0.f) ? 1.f : -1.f;   // sign-fold so inside test is w' >= 0
    float iz0 = 1.f / fmaxf(z0, EPSF);
    float iz1 = 1.f / fmaxf(z1, EPSF);
    float iz2 = 1.f / fmaxf(z2, EPSF);
    float Az = zok ? (A0 * iz0 + A1 * iz1 + A2 * iz2) / sa : 0.f;
    float Bz = zok ? (B0 * iz0 + B1 * iz1 + B2 * iz2) / sa : 0.f;
    float Cz = zok ? (C0 * iz0 + C1 * iz1 + C2 * iz2) / sa : 0.f;
    float pc = -BIGF;  // poison: invalid triangle -> edge value < 0 everywhere
    float* base = tri + (size_t)(n * NCHUNK + (f >> 4)) * 256 + (f & 15) * 4;
    base[0]   = s * A0; base[1]   = s * B0; base[2]   = zok ? s * C0 : pc; base[3]   = 0.f;
    base[64]  = s * A1; base[65]  = s * B1; base[66]  = zok ? s * C1 : pc; base[67]  = 0.f;
    base[128] = s * A2; base[129] = s * B2; base[130] = zok ? s * C2 : pc; base[131] = 0.f;
    base[192] = Az;     base[193] = Bz;     base[194] = Cz;                base[195] = 0.f;
}

// ---------------- K3: WMMA rasterizer ----------------
// Wave handles 16 pixels (one row segment) vs all triangles; 8 waves/WG share LDS tiles.
// Depth test runs on zi directly (max zi == min 1/zi) so the hot loop has NO divides.
__global__ void __launch_bounds__(256) k_raster(const float* __restrict__ tri,
                                                int* __restrict__ outI,
                                                float* __restrict__ outM) {
    __shared__ float lds[16 * 256];           // 16 chunks * 1KB
    int tid = threadIdx.x;
    int lane = tid & 31;
    int wave = tid >> 5;
    int n = blockIdx.z, y = blockIdx.y;
    int x0 = (blockIdx.x * 8 + wave) * 16;
    int m = lane & 15;
    bool hi = lane >= 16;
    // A-matrix 16x4: lanes 0-15 hold K=0,1 (px,py); lanes 16-31 hold K=2,3 (1,0)
    v2f a;
    a.x = hi ? 1.0f : (float)(x0 + m);
    a.y = hi ? 0.0f : (float)y;
    float bz[8]; int bi[8];
#pragma unroll
    for (int r = 0; r < 8; ++r) { bz[r] = EPSF; bi[r] = -1; }
    const float* trin = tri + (size_t)n * NCHUNK * 256;
    int boff = m * 4 + (hi ? 2 : 0);          // B-matrix lane offset (K pair select)
    unsigned ldsbase = (unsigned)(size_t)lds; // low 32 bits of flat LDS addr = LDS byte offset
    v8f zero = {0.f, 0.f, 0.f, 0.f, 0.f, 0.f, 0.f, 0.f};
    for (int cg = 0; cg < NCHUNK / 16; ++cg) {
        __syncthreads();
        const char* src = (const char*)(trin + cg * 16 * 256);
#pragma unroll
        for (int i = 0; i < 4; ++i) {
            unsigned loff = ldsbase + (unsigned)(tid + i * 256) * 16u;
            const char* g = src + (size_t)(tid + i * 256) * 16;
            // CDNA5 async global->LDS copy (tracked by ASYNCcnt, bypasses VGPRs)
            asm volatile("global_load_async_to_lds_b128 %0, %1, off"
                         :: "v"(loff), "v"(g) : "memory");
        }
#if __has_builtin(__builtin_amdgcn_s_wait_asynccnt)
        __builtin_amdgcn_s_wait_asynccnt(0);
#else
        asm volatile("s_wait_asynccnt 0x0" ::: "memory");
#endif
        __syncthreads();
        for (int c = 0; c < 16; ++c) {
            const float* cb = lds + c * 256 + boff;
            v2f b0; b0.x = cb[0];   b0.y = cb[1];
            v2f b1; b1.x = cb[64];  b1.y = cb[65];
            v2f b2; b2.x = cb[128]; b2.y = cb[129];
            v2f bzc; bzc.x = cb[192]; bzc.y = cb[193];
            v8f e0 = __builtin_amdgcn_wmma_f32_16x16x4_f32(false, a, false, b0,  (short)0, zero, false, false);
            v8f e1 = __builtin_amdgcn_wmma_f32_16x16x4_f32(false, a, false, b1,  (short)0, zero, false, false);
            v8f e2 = __builtin_amdgcn_wmma_f32_16x16x4_f32(false, a, false, b2,  (short)0, zero, false, false);
            v8f ez = __builtin_amdgcn_wmma_f32_16x16x4_f32(false, a, false, bzc, (short)0, zero, false, false);
            int tindex = (cg * 16 + c) * 16 + m;
#pragma unroll
            for (int r = 0; r < 8; ++r) {
                float m3 = fminf(fminf(e0[r], e1[r]), e2[r]);   // v_min3
                bool take = (m3 >= 0.f) && (ez[r] > bz[r]);     // strict >: earliest idx wins ties
                if (take) { bz[r] = ez[r]; bi[r] = tindex; }
            }
        }
    }
    // max-zi / argmax butterfly across 16 lanes (columns); lowest index wins ties
#pragma unroll
    for (int off = 1; off < 16; off <<= 1) {
#pragma unroll
        for (int r = 0; r < 8; ++r) {
            float z2 = __shfl_xor(bz[r], off, 32);
            int   i2 = __shfl_xor(bi[r], off, 32);
            if (z2 > bz[r] || (z2 == bz[r] && (unsigned)i2 < (unsigned)bi[r])) {
                bz[r] = z2; bi[r] = i2;
            }
        }
    }
    if (m == 0) {
        int rowbase = (n * HH + y) * WW + x0 + (hi ? 8 : 0);
#pragma unroll
        for (int r = 0; r < 8; ++r) {
            outI[rowbase + r] = bi[r];
            outM[rowbase + r] = (bi[r] >= 0) ? 1.0f : 0.0f;
        }
    }
}

// ---------------- K4: interpolation + border bilinear texture sample ----------------
__global__ void k_interp(const float4* __restrict__ vpix, const float* __restrict__ vt,
                         const int* __restrict__ vi, const int* __restrict__ vti,
                         const float* __restrict__ tex, const int* __restrict__ idxImg,
                         float* __restrict__ render, float* __restrict__ depth,
                         float* __restrict__ vtimg, float* __restrict__ bary) {
    int t = blockIdx.x * blockDim.x + threadIdx.x;
    if (t >= NIMG * HH * WW) return;
    int n = t / (HH * WW);
    int p = t % (HH * WW);
    int y = p / WW, x = p % WW;
    int idx = idxImg[t];
    float r0 = 0.f, r1 = 0.f, r2 = 0.f, dep = 0.f, uo = 0.f, vo = 0.f;
    float bo0 = 0.f, bo1 = 0.f, bo2 = 0.f;
    if (idx >= 0) {
        int f0 = vi[idx * 3 + 0], f1 = vi[idx * 3 + 1], f2 = vi[idx * 3 + 2];
        float4 p0 = vpix[n * NV + f0];
        float4 p1 = vpix[n * NV + f1];
        float4 p2 = vpix[n * NV + f2];
        float px = (float)x, py = (float)y;
        float w0 = (px - p1.x) * (p2.y - p1.y) - (py - p1.y) * (p2.x - p1.x);
        float w1 = (px - p2.x) * (p0.y - p2.y) - (py - p2.y) * (p0.x - p2.x);
        float w2 = (px - p0.x) * (p1.y - p0.y) - (py - p0.y) * (p1.x - p0.x);
        float area = (p1.x - p0.x) * (p2.y - p0.y) - (p1.y - p0.y) * (p2.x - p0.x);
        float sa = (fabsf(area) > EPSF) ? area : 1.0f;
        float z0 = fmaxf(p0.z, EPSF), z1 = fmaxf(p1.z, EPSF), z2 = fmaxf(p2.z, EPSF);
        float bz0 = w0 / sa / z0, bz1 = w1 / sa / z1, bz2 = w2 / sa / z2;
        float zi = fmaxf(bz0 + bz1 + bz2, EPSF);
        bo0 = bz0 / zi; bo1 = bz1 / zi; bo2 = bz2 / zi;
        dep = 1.0f / zi;
        int t0 = vti[idx * 3 + 0], t1 = vti[idx * 3 + 1], t2 = vti[idx * 3 + 2];
        float u0 = vt[t0 * 2 + 0], v0 = 1.0f - vt[t0 * 2 + 1];
        float u1 = vt[t1 * 2 + 0], v1 = 1.0f - vt[t1 * 2 + 1];
        float u2 = vt[t2 * 2 + 0], v2 = 1.0f - vt[t2 * 2 + 1];
        uo = bo0 * u0 + bo1 * u1 + bo2 * u2;
        vo = bo0 * v0 + bo1 * v1 + bo2 * v2;
        // grid = vt*2-1; sample with border clamp
        float gx = uo * 2.f - 1.f, gy = vo * 2.f - 1.f;
        float ix = (gx + 1.f) * 0.5f * (float)TEXW - 0.5f;
        float iy = (gy + 1.f) * 0.5f * (float)TEXW - 0.5f;
        ix = fminf(fmaxf(ix, -65536.f), 65536.f);
        iy = fminf(fmaxf(iy, -65536.f), 65536.f);
        float x0f = floorf(ix), y0f = floorf(iy);
        float fx = ix - x0f, fy = iy - y0f;
        int x0i = min(max((int)x0f, 0), TEXW - 1);
        int x1i = min(max((int)x0f + 1, 0), TEXW - 1);
        int y0i = min(max((int)y0f, 0), TEXW - 1);
        int y1i = min(max((int)y0f + 1, 0), TEXW - 1);
        const float* texn = tex + (size_t)n * 3 * TEXW * TEXW;
        float out3[3];
#pragma unroll
        for (int c = 0; c < 3; ++c) {
            const float* tc = texn + (size_t)c * TEXW * TEXW;
            float v00 = tc[y0i * TEXW + x0i];
            float v01 = tc[y0i * TEXW + x1i];
            float v10 = tc[y1i * TEXW + x0i];
            float v11 = tc[y1i * TEXW + x1i];
            out3[c] = (v00 * (1.f - fx) + v01 * fx) * (1.f - fy)
                    + (v10 * (1.f - fx) + v11 * fx) * fy;
        }
        r0 = out3[0]; r1 = out3[1]; r2 = out3[2];
    }
    int hw = HH * WW;
    render[(size_t)n * 3 * hw + 0 * hw + p] = r0;
    render[(size_t)n * 3 * hw + 1 * hw + p] = r1;
    render[(size_t)n * 3 * hw + 2 * hw + p] = r2;
    depth[t] = dep;
    vtimg[(size_t)t * 2 + 0] = uo;
    vtimg[(size_t)t * 2 + 1] = vo;
    bary[(size_t)n * 3 * hw + 0 * hw + p] = bo0;
    bary[(size_t)n * 3 * hw + 1 * hw + p] = bo1;
    bary[(size_t)n * 3 * hw + 2 * hw + p] = bo2;
}

extern "C" void kernel_launch(void* const* d_in, const int* in_sizes, int n_in,
                              void* d_out, int out_size, void* d_ws, size_t ws_size,
                              hipStream_t stream) {
    const float* v       = (const float*)d_in[0];
    const float* tex     = (const float*)d_in[1];
    const float* vt      = (const float*)d_in[2];
    const int*   vi      = (const int*)d_in[3];
    const int*   vti     = (const int*)d_in[4];
    const float* campos  = (const float*)d_in[5];
    const float* camrot  = (const float*)d_in[6];
    const float* focal   = (const float*)d_in[7];
    const float* princpt = (const float*)d_in[8];

    float* out    = (float*)d_out;
    float* render = out;                        // 2*3*512*512 = 1572864
    float* mask   = out + 1572864;              // 524288
    float* depth  = out + 2097152;              // 524288
    float* vtimg  = out + 2621440;              // 1048576
    float* bary   = out + 3670016;              // 1572864
    int*   idxImg = (int*)(out + 5242880);      // 524288 (int32 bits)

    float*  ws   = (float*)d_ws;
    float4* vpix = (float4*)ws;                 // NIMG*NV float4 = 32768 floats
    float*  tri  = ws + 32768;                  // NIMG*NCHUNK*256 = 65536 floats

    k_project<<<(NIMG * NV + 255) / 256, 256, 0, stream>>>(v, campos, camrot, focal, princpt, vpix);
    k_trisetup<<<(NIMG * NF + 255) / 256, 256, 0, stream>>>(vpix, vi, tri);
    dim3 g(WW / 128, HH, NIMG);
    k_raster<<<g, 256, 0, stream>>>(tri, idxImg, mask);
    k_interp<<<(NIMG * HH * WW + 255) / 256, 256, 0, stream>>>(vpix, vt, vi, vti, tex, idxImg,
                                                               render, depth, vtimg, bary);
}